// SOM_5634997092932
// MI455X (gfx1250) — compile-verified
//
#include <hip/hip_runtime.h>
#include <hip/hip_bf16.h>

typedef float v2f __attribute__((ext_vector_type(2)));
typedef float v8f __attribute__((ext_vector_type(8)));

#define N_PTS 65536
#define DIMS  64
#define KN    4096

// Precompute w2[k] = sum_d weight[d,k]^2 into workspace; zero the loss slot.
__global__ __launch_bounds__(256) void som_prep(const float* __restrict__ weight,
                                                float* __restrict__ w2,
                                                float* __restrict__ out) {
    const int k = blockIdx.x * 256 + threadIdx.x;
    if (blockIdx.x == 0 && threadIdx.x == 0) out[0] = 0.0f;
    if (k < KN) {
        float s = 0.0f;
#pragma unroll
        for (int d = 0; d < DIMS; ++d) {
            const float v = weight[d * KN + k];
            s = fmaf(v, v, s);
        }
        w2[k] = s;
    }
}

// Per-strip epilogue: reduce (min, idx) across the 16 lanes of each half,
// tie-break on smaller index (jnp.argmin semantics), write location of the
// winner, return this lane's distance contribution.
__device__ inline float strip_epilogue(float (&runMin)[8], int (&runIdx)[8],
                                       float x2c, int row0,
                                       const float* __restrict__ location,
                                       float* __restrict__ out, int lane) {
    const int half = lane >> 4;
    const int lrow = lane & 15;
#pragma unroll
    for (int i = 0; i < 8; ++i) {
        float m = runMin[i];
        int  id = runIdx[i];
#pragma unroll
        for (int off = 1; off < 16; off <<= 1) {
            const float om = __shfl_xor(m, off, 32);
            const int  oid = __shfl_xor(id, off, 32);
            if (om < m || (om == m && oid < id)) { m = om; id = oid; }
        }
        runMin[i] = m; runIdx[i] = id;
    }
    // Lane (lrow<8) handles strip-row r = (lrow&7) + 8*half.
    const int isel = lrow & 7;
    const int r = isel + half * 8;
    float mfin = runMin[0];
    int   kfin = runIdx[0];
#pragma unroll
    for (int i = 1; i < 8; ++i) {
        const bool p = (isel == i);
        mfin = p ? runMin[i] : mfin;
        kfin = p ? runIdx[i] : kfin;
    }
    const float xx = __shfl(x2c, r, 32);  // x2 of row r lives in lane r
    float contrib = 0.0f;
    if (lrow < 8) {
        const float dist = sqrtf(fmaxf(xx + mfin, 0.0f));
        const int   gr   = row0 + r;
        out[1 + 2 * gr]     = location[2 * kfin];
        out[1 + 2 * gr + 1] = location[2 * kfin + 1];
        contrib = dist;
    }
    return contrib;
}

// One wave per TWO 16-row strips (32 rows): each B tile (32 coalesced b32
// loads) feeds 32 chained V_WMMA_F32_16X16X4_F32 across two independent
// accumulator chains -> 1:1 load:WMMA ratio, half the weight re-read traffic.
__global__ __launch_bounds__(256) void som_main(const float* __restrict__ x,
                                                const float* __restrict__ weight,
                                                const float* __restrict__ location,
                                                const float* __restrict__ w2,
                                                float* __restrict__ out) {
    const int lane = threadIdx.x & 31;
    const int wave = threadIdx.x >> 5;
    const int half = lane >> 4;   // which 16-lane half
    const int lrow = lane & 15;
    const int row0 = blockIdx.x * 256 + wave * 32;  // strips: row0, row0+16

    // ---- A strips in WMMA A-layout: lane holds row lrow, K-slots
    //      2*half + {0,1} of each K=4 chunk. Aligned b64 loads.
    v2f a0[16], a1[16];
    const float* xr0 = x + (row0 + lrow) * DIMS + half * 2;
    const float* xr1 = xr0 + 16 * DIMS;
#pragma unroll
    for (int t = 0; t < 16; ++t) {
        a0[t] = *(const v2f*)(xr0 + t * 4);
        a1[t] = *(const v2f*)(xr1 + t * 4);
    }

    // x2 per row of each strip: lane partial + partner half.
    float x2p0 = 0.0f, x2p1 = 0.0f;
#pragma unroll
    for (int t = 0; t < 16; ++t) {
        x2p0 = fmaf(a0[t].x, a0[t].x, fmaf(a0[t].y, a0[t].y, x2p0));
        x2p1 = fmaf(a1[t].x, a1[t].x, fmaf(a1[t].y, a1[t].y, x2p1));
    }
    const float x2c0 = x2p0 + __shfl_xor(x2p0, 16, 32);
    const float x2c1 = x2p1 + __shfl_xor(x2p1, 16, 32);

    float runMin0[8], runMin1[8];
    int   runIdx0[8], runIdx1[8];
#pragma unroll
    for (int i = 0; i < 8; ++i) {
        runMin0[i] = 3.4e38f; runIdx0[i] = 0;
        runMin1[i] = 3.4e38f; runIdx1[i] = 0;
    }

    for (int kt = 0; kt < KN / 16; ++kt) {
        const int col = kt * 16 + lrow;
        const float* wp = weight + (half * 2) * KN + col;

        // B tile in WMMA B-layout: lane holds col lrow, K-slots 2*half+{0,1}.
        v2f b[16];
#pragma unroll
        for (int t = 0; t < 16; ++t) {
            b[t].x = wp[t * 4 * KN];
            b[t].y = wp[t * 4 * KN + KN];
        }
        if (kt + 1 < KN / 16) __builtin_prefetch(wp + 16, 0, 0);

        v8f c0 = {}, c1 = {};
#pragma unroll
        for (int t = 0; t < 16; ++t) {
            c0 = __builtin_amdgcn_wmma_f32_16x16x4_f32(
                     false, a0[t], false, b[t], (short)0, c0, false, false);
            c1 = __builtin_amdgcn_wmma_f32_16x16x4_f32(
                     false, a1[t], false, b[t], (short)0, c1, false, false);
        }

        // C layout: VGPR i, this lane -> row (i + 8*half), col lrow of tile.
        const float w2k = w2[col];
#pragma unroll
        for (int i = 0; i < 8; ++i) {
            const float s0 = fmaf(-2.0f, c0[i], w2k);
            if (s0 < runMin0[i]) { runMin0[i] = s0; runIdx0[i] = col; }
            const float s1 = fmaf(-2.0f, c1[i], w2k);
            if (s1 < runMin1[i]) { runMin1[i] = s1; runIdx1[i] = col; }
        }
    }

    float contrib = strip_epilogue(runMin0, runIdx0, x2c0, row0,      location, out, lane);
    contrib      += strip_epilogue(runMin1, runIdx1, x2c1, row0 + 16, location, out, lane);

#pragma unroll
    for (int off = 1; off < 32; off <<= 1)
        contrib += __shfl_xor(contrib, off, 32);
    if (lane == 0) atomicAdd(out, contrib * (1.0f / (float)N_PTS));
}

extern "C" void kernel_launch(void* const* d_in, const int* in_sizes, int n_in,
                              void* d_out, int out_size, void* d_ws, size_t ws_size,
                              hipStream_t stream) {
    const float* x        = (const float*)d_in[0];   // [65536, 64]
    const float* weight   = (const float*)d_in[1];   // [64, 4096]
    const float* location = (const float*)d_in[2];   // [4096, 2]
    float* out = (float*)d_out;                      // [1 + 65536*2]
    float* w2  = (float*)d_ws;                       // 4096 floats scratch

    som_prep<<<KN / 256, 256, 0, stream>>>(weight, w2, out);
    som_main<<<N_PTS / 256, 256, 0, stream>>>(x, weight, location, w2, out);
}